// Node2VecHypergraphConv_60189671686410
// MI455X (gfx1250) — compile-verified
//
#include <hip/hip_runtime.h>

#define N_NODES  50000
#define N_HEDGES 10000
#define N_INC    800000
#define CDIM     128
#define LSTRIDE  132   // LDS row stride (floats): 132 mod 64 = 4 -> conflict-free b64 reads

typedef float v2f __attribute__((ext_vector_type(2)));
typedef float v8f __attribute__((ext_vector_type(8)));

// ---------------------------------------------------------------------------
// Degree / hyperedge-size counting: D[node] += 1, B[hedge] += 1
// ---------------------------------------------------------------------------
__global__ __launch_bounds__(256) void degree_kernel(
    const int* __restrict__ node_idx, const int* __restrict__ hedge_idx,
    float* __restrict__ Dcnt, float* __restrict__ Bcnt) {
  int i = blockIdx.x * blockDim.x + threadIdx.x;
  if (i < N_INC) {
    atomicAdd(&Dcnt[node_idx[i]], 1.0f);
    atomicAdd(&Bcnt[hedge_idx[i]], 1.0f);
  }
}

// ---------------------------------------------------------------------------
// GEMM1: x = emb @ W_hg^T   (f32 WMMA 16x16x4, one 16x128 tile per wave)
// ---------------------------------------------------------------------------
__global__ __launch_bounds__(256) void gemm_xw_kernel(
    const float* __restrict__ A,   // [N_NODES, 128]
    const float* __restrict__ W,   // [128, 128] row-major (out_c, in_k)
    float* __restrict__ out) {     // [N_NODES, 128]
  __shared__ float sW[CDIM * LSTRIDE];
  for (int i = threadIdx.x; i < CDIM * CDIM; i += 256) {
    int r = i >> 7, c = i & 127;
    sW[r * LSTRIDE + c] = W[i];
  }
  __syncthreads();

  int wave = threadIdx.x >> 5;
  int lane = threadIdx.x & 31;
  int m0 = (blockIdx.x * 8 + wave) * 16;
  if (m0 >= N_NODES) return;               // wave-uniform; EXEC stays all-ones

  int col = lane & 15;                     // A: M index / B,D: N index
  int hi  = lane >> 4;                     // 0: K={k,k+1}  1: K={k+2,k+3}

  v8f acc[8];
  #pragma unroll
  for (int ct = 0; ct < 8; ++ct)
    #pragma unroll
    for (int j = 0; j < 8; ++j) acc[ct][j] = 0.0f;

  const float* arow = A + (size_t)(m0 + col) * CDIM;
  for (int k = 0; k < CDIM; k += 4) {
    int kk = k + 2 * hi;
    v2f a = *(const v2f*)(arow + kk);      // A[m0+col][kk..kk+1]
    #pragma unroll
    for (int ct = 0; ct < 8; ++ct) {
      v2f b = *(const v2f*)(&sW[(ct * 16 + col) * LSTRIDE + kk]); // W[ct*16+col][kk..]
      acc[ct] = __builtin_amdgcn_wmma_f32_16x16x4_f32(
          false, a, false, b, (short)0, acc[ct], false, false);
    }
  }

  #pragma unroll
  for (int ct = 0; ct < 8; ++ct)
    #pragma unroll
    for (int j = 0; j < 8; ++j)
      out[(size_t)(m0 + j + 8 * hi) * CDIM + ct * 16 + col] = acc[ct][j];
}

// ---------------------------------------------------------------------------
// Pass 1: e[hedge] += x[node]   (one wave per incidence, float4 per lane)
// ---------------------------------------------------------------------------
__global__ __launch_bounds__(256) void scatter1_kernel(
    const int* __restrict__ node_idx, const int* __restrict__ hedge_idx,
    const float* __restrict__ x, float* __restrict__ e) {
  int gw   = (blockIdx.x * blockDim.x + threadIdx.x) >> 5;
  int lane = threadIdx.x & 31;
  if (gw >= N_INC) return;
  int n = node_idx[gw], h = hedge_idx[gw];
  const float4 v = *(const float4*)(x + (size_t)n * CDIM + lane * 4);
  float* ep = e + (size_t)h * CDIM + lane * 4;
  atomicAdd(ep + 0, v.x);
  atomicAdd(ep + 1, v.y);
  atomicAdd(ep + 2, v.z);
  atomicAdd(ep + 3, v.w);
}

// ---------------------------------------------------------------------------
// Pass 2: acc[node] += e[hedge] * B^-1[hedge]
// ---------------------------------------------------------------------------
__global__ __launch_bounds__(256) void scatter2_kernel(
    const int* __restrict__ node_idx, const int* __restrict__ hedge_idx,
    const float* __restrict__ e, const float* __restrict__ Bcnt,
    float* __restrict__ acc) {
  int gw   = (blockIdx.x * blockDim.x + threadIdx.x) >> 5;
  int lane = threadIdx.x & 31;
  if (gw >= N_INC) return;
  int n = node_idx[gw], h = hedge_idx[gw];
  float bc   = Bcnt[h];
  float binv = bc > 0.0f ? 1.0f / bc : 0.0f;
  const float4 v = *(const float4*)(e + (size_t)h * CDIM + lane * 4);
  float* ap = acc + (size_t)n * CDIM + lane * 4;
  atomicAdd(ap + 0, v.x * binv);
  atomicAdd(ap + 1, v.y * binv);
  atomicAdd(ap + 2, v.z * binv);
  atomicAdd(ap + 3, v.w * binv);
}

// ---------------------------------------------------------------------------
// GEMM2: out = leaky_relu(acc * D^-1 + b_hg) @ W_lin^T + b_lin
// (normalization/bias/activation fused into the A-fragment load,
//  b_lin fused into the C accumulator init)
// ---------------------------------------------------------------------------
__global__ __launch_bounds__(256) void gemm_out_kernel(
    const float* __restrict__ Agg,   // [N_NODES, 128] raw node accumulator
    const float* __restrict__ Dcnt,  // [N_NODES]
    const float* __restrict__ b_hg,  // [128]
    const float* __restrict__ W,     // W_lin [128, 128]
    const float* __restrict__ b_lin, // [128]
    float* __restrict__ out) {       // [N_NODES, 128]
  __shared__ float sW[CDIM * LSTRIDE];
  for (int i = threadIdx.x; i < CDIM * CDIM; i += 256) {
    int r = i >> 7, c = i & 127;
    sW[r * LSTRIDE + c] = W[i];
  }
  __syncthreads();

  int wave = threadIdx.x >> 5;
  int lane = threadIdx.x & 31;
  int m0 = (blockIdx.x * 8 + wave) * 16;
  if (m0 >= N_NODES) return;

  int col = lane & 15;
  int hi  = lane >> 4;

  v8f acc[8];
  #pragma unroll
  for (int ct = 0; ct < 8; ++ct) {
    float bv = b_lin[ct * 16 + col];       // same output column for all 8 rows
    #pragma unroll
    for (int j = 0; j < 8; ++j) acc[ct][j] = bv;
  }

  float dc   = Dcnt[m0 + col];
  float dinv = dc > 0.0f ? 1.0f / dc : 0.0f;

  const float* arow = Agg + (size_t)(m0 + col) * CDIM;
  for (int k = 0; k < CDIM; k += 4) {
    int kk = k + 2 * hi;
    float a0 = arow[kk + 0] * dinv + b_hg[kk + 0];
    float a1 = arow[kk + 1] * dinv + b_hg[kk + 1];
    a0 = a0 > 0.0f ? a0 : a0 * 0.01f;      // leaky_relu, slope 0.01
    a1 = a1 > 0.0f ? a1 : a1 * 0.01f;
    v2f a; a[0] = a0; a[1] = a1;
    #pragma unroll
    for (int ct = 0; ct < 8; ++ct) {
      v2f b = *(const v2f*)(&sW[(ct * 16 + col) * LSTRIDE + kk]);
      acc[ct] = __builtin_amdgcn_wmma_f32_16x16x4_f32(
          false, a, false, b, (short)0, acc[ct], false, false);
    }
  }

  #pragma unroll
  for (int ct = 0; ct < 8; ++ct)
    #pragma unroll
    for (int j = 0; j < 8; ++j)
      out[(size_t)(m0 + j + 8 * hi) * CDIM + ct * 16 + col] = acc[ct][j];
}

// ---------------------------------------------------------------------------
extern "C" void kernel_launch(void* const* d_in, const int* in_sizes, int n_in,
                              void* d_out, int out_size, void* d_ws, size_t ws_size,
                              hipStream_t stream) {
  const int*   node_idx  = (const int*)d_in[0];
  const int*   hedge_idx = (const int*)d_in[1];
  const float* emb       = (const float*)d_in[2];
  const float* W_hg      = (const float*)d_in[3];
  const float* b_hg      = (const float*)d_in[4];
  const float* W_lin     = (const float*)d_in[5];
  const float* b_lin     = (const float*)d_in[6];
  float*       out       = (float*)d_out;

  // workspace layout (bytes):
  //   [0, 25.6M)      x  (node features)  -- reused as node accumulator
  //   [25.6M, 30.72M) e  (hyperedge sums)
  //   [30.72M, 30.92M) Dcnt (node degrees)
  //   [30.92M, 30.96M) Bcnt (hyperedge sizes)
  char*  ws   = (char*)d_ws;
  float* x    = (float*)(ws);
  float* e    = (float*)(ws + 25600000);
  float* Dcnt = (float*)(ws + 30720000);
  float* Bcnt = (float*)(ws + 30920000);

  hipMemsetAsync(Dcnt, 0, (size_t)(N_NODES + N_HEDGES) * sizeof(float), stream);
  hipMemsetAsync(e,    0, (size_t)N_HEDGES * CDIM * sizeof(float), stream);

  degree_kernel<<<(N_INC + 255) / 256, 256, 0, stream>>>(node_idx, hedge_idx, Dcnt, Bcnt);

  // x = emb @ W_hg^T : 3125 row-tiles, 8 waves/block
  gemm_xw_kernel<<<391, 256, 0, stream>>>(emb, W_hg, x);

  // one wave per incidence
  scatter1_kernel<<<(N_INC * 32) / 256, 256, 0, stream>>>(node_idx, hedge_idx, x, e);

  // recycle x as the node accumulator (stream order guarantees pass1 is done)
  hipMemsetAsync(x, 0, (size_t)N_NODES * CDIM * sizeof(float), stream);

  scatter2_kernel<<<(N_INC * 32) / 256, 256, 0, stream>>>(node_idx, hedge_idx, e, Bcnt, x);

  gemm_out_kernel<<<391, 256, 0, stream>>>(x, Dcnt, b_hg, W_lin, b_lin, out);
}